// Informer_59141699666539
// MI455X (gfx1250) — compile-verified
//
#include <hip/hip_runtime.h>
#include <hip/hip_bf16.h>
#include <math.h>
#include <stdint.h>

// ---------------- problem constants (from reference) ----------------
#define BATCH   4
#define SEQ     2048
#define D_IN    64
#define DMODEL  512
#define NHEAD   8
#define DK      64
#define NLAYER  2
#define NCLS    2
#define TOPK    35
#define LN_EPS  1e-5f

typedef __attribute__((ext_vector_type(2))) float        v2f;
typedef __attribute__((ext_vector_type(8))) float        v8f;
typedef __attribute__((ext_vector_type(4))) unsigned int u32x4;
typedef __attribute__((ext_vector_type(8))) int          i32x8;
typedef __attribute__((ext_vector_type(4))) int          i32x4;

// Tensor Data Mover available? (device pass on gfx1250)
#if defined(__gfx1250__) && __has_builtin(__builtin_amdgcn_tensor_load_to_lds)
#define HAVE_TDM 1
#else
#define HAVE_TDM 0
#endif

#if HAVE_TDM
// ---------------------------------------------------------------------
// Issue one TDM 2D tile load: global (row-major, row_stride elems) -> LDS.
// Tile is tile_rows x tile_cols f32; rem_rows/rem_cols = distance from the
// tile origin to the tensor edge (TDM zero-fills OOB reads).
// pad_code: LDS pad interval code (4 => every 32 DWORDs, 5 => every 64),
// pad amount fixed at 1 DWORD (bank-conflict padding).
// Descriptor packing per CDNA5 ISA.
// This toolchain exposes the 6-arg builtin:
//   (uint32x4 g0, int32x8 g1, int32x4 g2, int32x4 g3, int32x8, i32 cpol)
// ---------------------------------------------------------------------
__device__ __forceinline__ void tdm_load_tile_f32(
    uint32_t lds_byte_addr, const float* tile_ptr,
    uint32_t rem_cols, uint32_t rem_rows,
    uint32_t tile_cols, uint32_t tile_rows,
    uint64_t row_stride_elems, uint32_t pad_code)
{
    uint64_t ga = (uint64_t)(uintptr_t)tile_ptr;
    u32x4 g0;
    g0[0] = 1u;                                        // count=1, user desc
    g0[1] = lds_byte_addr;                             // lds_addr
    g0[2] = (uint32_t)ga;                              // global_addr[31:0]
    g0[3] = (uint32_t)((ga >> 32) & 0x01FFFFFFu)       // global_addr[56:32]
          | (2u << 30);                                // type = 2 ("image")
    i32x8 g1;
    g1[0] = (int)((2u << 16)                           // data_size = 4B
                | (1u << 20)                           // pad_enable
                | (pad_code << 22));                   // pad_interval (amount=1 DW)
    g1[1] = (int)((rem_cols & 0xFFFFu) << 16);         // tensor_dim0[15:0]
    g1[2] = (int)(((rem_cols >> 16) & 0xFFFFu)         // tensor_dim0[31:16]
                | ((rem_rows & 0xFFFFu) << 16));       // tensor_dim1[15:0]
    g1[3] = (int)(((rem_rows >> 16) & 0xFFFFu)         // tensor_dim1[31:16]
                | (tile_cols << 16));                  // tile_dim0
    g1[4] = (int)(tile_rows & 0xFFFFu);                // tile_dim1 (tile_dim2=0)
    g1[5] = (int)(row_stride_elems & 0xFFFFFFFFull);   // dim0_stride[31:0]
    g1[6] = (int)((row_stride_elems >> 32) & 0xFFFFull); // dim0_stride[47:32]
    g1[7] = 0;
    i32x4 z4 = {0, 0, 0, 0};                           // 2D: groups 2/3 unused
    i32x8 z8 = {0, 0, 0, 0, 0, 0, 0, 0};
    __builtin_amdgcn_tensor_load_to_lds(g0, g1, z4, z4, z8, 0);
}
#endif

// =====================================================================
// Generic fp32 WMMA GEMM:  C[M,N] = act(A[M,K] @ W[K,N] + bias[N] [+PE])
// Block: 256 threads = 8 waves; C-tile 64x32; each wave one 16x16 tile.
// flags: bit0 = ReLU, bit1 = add sinusoidal positional encoding
// TDM path: double-buffered LDS tiles, DMA overlapped with WMMA compute.
// =====================================================================
__global__ __launch_bounds__(256)
void gemm_bias_kernel(const float* __restrict__ A,
                      const float* __restrict__ W,
                      const float* __restrict__ bias,
                      float* __restrict__ C,
                      int M, int N, int K, int flags, int Sdim, int Dmod)
{
#if HAVE_TDM
    __shared__ float a_t[2][64][65];   // +1 pad via TDM pad_interval=64,amount=1
    __shared__ float b_t[2][64][33];   // +1 pad via TDM pad_interval=32,amount=1
#else
    __shared__ float a_t[1][64][65];
    __shared__ float b_t[1][64][33];
#endif

    const int tid  = threadIdx.x;
    const int m0   = blockIdx.x * 64;
    const int n0   = blockIdx.y * 32;
    const int lane = tid & 31;
    const int wave = tid >> 5;
    const int mBase = (wave >> 1) * 16;   // 4 wave-rows
    const int nBase = (wave & 1) * 16;    // 2 wave-cols
    const int half = lane >> 4;
    const int l15  = lane & 15;

    v8f acc = {};

#if HAVE_TDM
    int cur = 0;
    if (tid == 0) {   // single wave issues DMA; EXEC ignored by tensor ops
        tdm_load_tile_f32((uint32_t)(uintptr_t)&a_t[0][0][0],
                          A + (size_t)m0 * K,
                          (uint32_t)K, (uint32_t)(M - m0), 64u, 64u,
                          (uint64_t)K, 5u);
        tdm_load_tile_f32((uint32_t)(uintptr_t)&b_t[0][0][0],
                          W + (size_t)0 * N + n0,
                          (uint32_t)(N - n0), (uint32_t)K, 32u, 64u,
                          (uint64_t)N, 4u);
    }
    for (int kb = 0; kb < K; kb += 64) {
        if (tid == 0) __builtin_amdgcn_s_wait_tensorcnt(0);
        __syncthreads();                       // buf[cur] tiles visible to all
        const int nxt = kb + 64;
        if (nxt < K && tid == 0) {             // prefetch next K-slab via TDM
            tdm_load_tile_f32((uint32_t)(uintptr_t)&a_t[cur ^ 1][0][0],
                              A + (size_t)m0 * K + nxt,
                              (uint32_t)(K - nxt), (uint32_t)(M - m0), 64u, 64u,
                              (uint64_t)K, 5u);
            tdm_load_tile_f32((uint32_t)(uintptr_t)&b_t[cur ^ 1][0][0],
                              W + (size_t)nxt * N + n0,
                              (uint32_t)(N - n0), (uint32_t)(K - nxt), 32u, 64u,
                              (uint64_t)N, 4u);
        }
        #pragma unroll
        for (int k4 = 0; k4 < 64; k4 += 4) {
            const int kk = k4 + 2 * half;      // A 16x4 / B 4x16 f32 layout
            v2f av, bv;
            av.x = a_t[cur][mBase + l15][kk];
            av.y = a_t[cur][mBase + l15][kk + 1];
            bv.x = b_t[cur][kk][nBase + l15];
            bv.y = b_t[cur][kk + 1][nBase + l15];
            acc = __builtin_amdgcn_wmma_f32_16x16x4_f32(
                      false, av, false, bv, (short)0, acc, false, false);
        }
        __syncthreads();                       // done reading buf[cur]
        cur ^= 1;
    }
#else
    for (int kb = 0; kb < K; kb += 64) {
        for (int i = tid; i < 64 * 64; i += 256) {
            int r = i >> 6, c = i & 63;
            int gr = m0 + r, gc = kb + c;
            a_t[0][r][c] = (gr < M && gc < K) ? A[(size_t)gr * K + gc] : 0.0f;
        }
        for (int i = tid; i < 64 * 32; i += 256) {
            int r = i >> 5, c = i & 31;
            int gr = kb + r, gc = n0 + c;
            b_t[0][r][c] = (gr < K && gc < N) ? W[(size_t)gr * N + gc] : 0.0f;
        }
        __syncthreads();
        #pragma unroll
        for (int k4 = 0; k4 < 64; k4 += 4) {
            const int kk = k4 + 2 * half;
            v2f av, bv;
            av.x = a_t[0][mBase + l15][kk];
            av.y = a_t[0][mBase + l15][kk + 1];
            bv.x = b_t[0][kk][nBase + l15];
            bv.y = b_t[0][kk + 1][nBase + l15];
            acc = __builtin_amdgcn_wmma_f32_16x16x4_f32(
                      false, av, false, bv, (short)0, acc, false, false);
        }
        __syncthreads();
    }
#endif

    #pragma unroll
    for (int vr = 0; vr < 8; ++vr) {
        int m = m0 + mBase + vr + 8 * half;    // C layout: VGPR r -> rows r, r+8
        int n = n0 + nBase + l15;
        if (m < M && n < N) {
            float val = acc[vr] + bias[n];
            if (flags & 2) {                   // sinusoidal PE (embedding only)
                int   s   = m % Sdim;
                int   i2  = n & ~1;
                float dv  = expf(-(float)i2 * (logf(10000.0f) / (float)Dmod));
                float ang = (float)s * dv;
                val += (n & 1) ? cosf(ang) : sinf(ang);
            }
            if (flags & 1) val = fmaxf(val, 0.0f);
            C[(size_t)m * N + n] = val;
        }
    }
}

// =====================================================================
// Fused sparse attention: per (b, head, 32-query block)
//   scores = q @ k^T * 1/sqrt(DK)   (WMMA, 64-key chunks)
//   streaming sorted top-35 per row -> softmax -> weights @ v[0:35]
// (faithful to reference quirk: indices discarded; weight rank j pairs
//  with v sequence row j)
// =====================================================================
__global__ __launch_bounds__(256)
void attn_topk_kernel(const float* __restrict__ q,
                      const float* __restrict__ k,
                      const float* __restrict__ v,
                      float* __restrict__ ao,
                      int Sn, int Dm)
{
    const int qblk = blockIdx.x;   // S/32 blocks
    const int hh   = blockIdx.y;   // head
    const int bb   = blockIdx.z;   // batch
    const int tid  = threadIdx.x;
    const int lane = tid & 31;
    const int wave = tid >> 5;
    const int off  = hh * DK;
    const int q0   = qblk * 32;
    const float scale = 0.125f;    // 1/sqrt(64)

    __shared__ float q_lds[32][DK + 1];
    __shared__ float kT[DK][64 + 1];
    __shared__ float sc[32][64 + 1];
    __shared__ float vt[TOPK][DK];
    __shared__ float topk[32][TOPK];   // sorted desc; reused for softmax weights

    for (int i = tid; i < 32 * DK; i += 256) {
        int r = i >> 6, c = i & 63;
        q_lds[r][c] = q[((size_t)(bb * Sn + q0 + r)) * Dm + off + c];
    }
    for (int i = tid; i < TOPK * DK; i += 256) {
        int r = i >> 6, c = i & 63;
        vt[r][c] = v[((size_t)(bb * Sn + r)) * Dm + off + c];
    }
    for (int i = tid; i < 32 * TOPK; i += 256)
        topk[i / TOPK][i % TOPK] = -3.0e38f;
    __syncthreads();

    const int rowBase = (wave >> 2) * 16;   // 2 wave-rows
    const int colBase = (wave & 3) * 16;    // 4 wave-cols
    const int half = lane >> 4;
    const int l15  = lane & 15;

    for (int ch = 0; ch < Sn; ch += 64) {
        // stage k^T chunk: kT[dim][key]  (transpose load; manual staging)
        for (int i = tid; i < 64 * DK; i += 256) {
            int kk = i & 63, j = i >> 6;
            kT[kk][j] = k[((size_t)(bb * Sn + ch + j)) * Dm + off + kk];
        }
        __syncthreads();

        v8f acc = {};
        #pragma unroll
        for (int k4 = 0; k4 < DK; k4 += 4) {
            const int kk = k4 + 2 * half;
            v2f av, bv;
            av.x = q_lds[rowBase + l15][kk];
            av.y = q_lds[rowBase + l15][kk + 1];
            bv.x = kT[kk][colBase + l15];
            bv.y = kT[kk + 1][colBase + l15];
            acc = __builtin_amdgcn_wmma_f32_16x16x4_f32(
                      false, av, false, bv, (short)0, acc, false, false);
        }
        #pragma unroll
        for (int vr = 0; vr < 8; ++vr)
            sc[rowBase + vr + 8 * half][colBase + l15] = acc[vr] * scale;
        __syncthreads();

        // streaming sorted top-35 update; one owner thread per query row
        if (tid < 32) {
            float* L = topk[tid];
            #pragma unroll 1
            for (int j = 0; j < 64; ++j) {
                float val = sc[tid][j];
                if (val > L[TOPK - 1]) {
                    int i = TOPK - 1;
                    while (i > 0 && L[i - 1] < val) { L[i] = L[i - 1]; --i; }
                    L[i] = val;
                }
            }
        }
        __syncthreads();
    }

    // softmax over sorted top-35 -> weights in place
    if (tid < 32) {
        float* L = topk[tid];
        float m = L[0], s = 0.0f;
        #pragma unroll 1
        for (int j = 0; j < TOPK; ++j) s += expf(L[j] - m);
        float inv = 1.0f / s;
        #pragma unroll 1
        for (int j = 0; j < TOPK; ++j) L[j] = expf(L[j] - m) * inv;
    }
    __syncthreads();

    // ao[row][d] = sum_j w[row][j] * v[j][d]   (first TOPK sequence rows)
    for (int i = tid; i < 32 * DK; i += 256) {
        int r = i >> 6, d = i & 63;
        float a2 = 0.0f;
        #pragma unroll 1
        for (int j = 0; j < TOPK; ++j) a2 += topk[r][j] * vt[j][d];
        ao[((size_t)(bb * Sn + q0 + r)) * Dm + off + d] = a2;
    }
}

// =====================================================================
// out[row] = LayerNorm(base[row] + delta[row]) * g + b   (block per row)
// =====================================================================
__global__ __launch_bounds__(256)
void ln_residual_kernel(const float* __restrict__ base,
                        const float* __restrict__ delta,
                        const float* __restrict__ g,
                        const float* __restrict__ b,
                        float* __restrict__ out, int Dm)
{
    const int row = blockIdx.x;
    const int tid = threadIdx.x;
    __shared__ float r1[256];
    __shared__ float r2[256];

    float s = 0.0f, sq = 0.0f;
    for (int d = tid; d < Dm; d += 256) {
        float x = base[(size_t)row * Dm + d] + delta[(size_t)row * Dm + d];
        s += x; sq += x * x;
    }
    r1[tid] = s; r2[tid] = sq;
    __syncthreads();
    for (int st = 128; st > 0; st >>= 1) {
        if (tid < st) { r1[tid] += r1[tid + st]; r2[tid] += r2[tid + st]; }
        __syncthreads();
    }
    float mean = r1[0] / (float)Dm;
    float var  = r2[0] / (float)Dm - mean * mean;
    float inv  = rsqrtf(var + LN_EPS);
    for (int d = tid; d < Dm; d += 256) {
        float x = base[(size_t)row * Dm + d] + delta[(size_t)row * Dm + d];
        out[(size_t)row * Dm + d] = (x - mean) * inv * g[d] + b[d];
    }
}

// =====================================================================
// pooled[b][d] = mean over s of h[b][s][d]
// =====================================================================
__global__ void pool_kernel(const float* __restrict__ h,
                            float* __restrict__ pooled,
                            int Bn, int Sn, int Dm)
{
    int idx = blockIdx.x * blockDim.x + threadIdx.x;
    if (idx >= Bn * Dm) return;
    int bb = idx / Dm, d = idx % Dm;
    const float* p = h + (size_t)bb * Sn * Dm + d;
    float s = 0.0f;
    for (int ss = 0; ss < Sn; ++ss) s += p[(size_t)ss * Dm];
    pooled[idx] = s / (float)Sn;
}

// =====================================================================
extern "C" void kernel_launch(void* const* d_in, const int* in_sizes, int n_in,
                              void* d_out, int out_size, void* d_ws, size_t ws_size,
                              hipStream_t stream)
{
    const float* x      = (const float*)d_in[0];
    const float* emb_w  = (const float*)d_in[1];
    const float* emb_b  = (const float*)d_in[2];
    const float* wq     = (const float*)d_in[3];
    const float* bq     = (const float*)d_in[4];
    const float* wk     = (const float*)d_in[5];
    const float* bk     = (const float*)d_in[6];
    const float* wv     = (const float*)d_in[7];
    const float* bv     = (const float*)d_in[8];
    const float* wo     = (const float*)d_in[9];
    const float* bo     = (const float*)d_in[10];
    const float* ff1_w  = (const float*)d_in[11];
    const float* ff1_b  = (const float*)d_in[12];
    const float* ff2_w  = (const float*)d_in[13];
    const float* ff2_b  = (const float*)d_in[14];
    const float* ln1_g  = (const float*)d_in[15];
    const float* ln1_b  = (const float*)d_in[16];
    const float* ln2_g  = (const float*)d_in[17];
    const float* ln2_b  = (const float*)d_in[18];
    const float* dec1_w = (const float*)d_in[19];
    const float* dec1_b = (const float*)d_in[20];
    const float* dec2_w = (const float*)d_in[21];
    const float* dec2_b = (const float*)d_in[22];

    const int M = BATCH * SEQ;                    // 8192
    const size_t TOK = (size_t)M * DMODEL;        // 4M floats

    float* ws     = (float*)d_ws;
    float* H      = ws;                           // [0, 4M)
    float* QB     = ws + TOK;                     // [4M, 8M)
    float* KB     = ws + 2 * TOK;                 // [8M, 12M)
    float* VB     = ws + 3 * TOK;                 // [12M, 16M)
    float* AOB    = ws + 4 * TOK;                 // [16M, 20M)
    float* TMPB   = ws + 5 * TOK;                 // [20M, 24M)
    float* FFB    = ws + TOK;                     // reuses QB..AOB (16M floats)
    float* POOLED = ws + 6 * TOK;                 // 4*512
    float* T1     = POOLED + BATCH * DMODEL;      // 4*256

    auto gemm = [&](const float* A, const float* W, const float* bias,
                    float* C, int Mm, int Nn, int Kk, int flags) {
        dim3 g((Mm + 63) / 64, (Nn + 31) / 32);
        gemm_bias_kernel<<<g, 256, 0, stream>>>(A, W, bias, C, Mm, Nn, Kk,
                                                flags, SEQ, DMODEL);
    };

    // embedding + positional encoding
    gemm(x, emb_w, emb_b, H, M, DMODEL, D_IN, /*PE*/2);

    for (int l = 0; l < NLAYER; ++l) {
        const size_t wofs = (size_t)l * DMODEL * DMODEL;
        const size_t bofs = (size_t)l * DMODEL;
        const size_t f1w  = (size_t)l * DMODEL * (4 * DMODEL);
        const size_t f1b  = (size_t)l * (4 * DMODEL);

        gemm(H, wq + wofs, bq + bofs, QB, M, DMODEL, DMODEL, 0);
        gemm(H, wk + wofs, bk + bofs, KB, M, DMODEL, DMODEL, 0);
        gemm(H, wv + wofs, bv + bofs, VB, M, DMODEL, DMODEL, 0);

        dim3 ag(SEQ / 32, NHEAD, BATCH);
        attn_topk_kernel<<<ag, 256, 0, stream>>>(QB, KB, VB, AOB, SEQ, DMODEL);

        gemm(AOB, wo + wofs, bo + bofs, TMPB, M, DMODEL, DMODEL, 0);
        ln_residual_kernel<<<M, 256, 0, stream>>>(H, TMPB, ln1_g + bofs,
                                                  ln1_b + bofs, H, DMODEL);

        gemm(H, ff1_w + f1w, ff1_b + f1b, FFB, M, 4 * DMODEL, DMODEL, /*ReLU*/1);
        gemm(FFB, ff2_w + f1w, ff2_b + bofs, TMPB, M, DMODEL, 4 * DMODEL, 0);
        ln_residual_kernel<<<M, 256, 0, stream>>>(H, TMPB, ln2_g + bofs,
                                                  ln2_b + bofs, H, DMODEL);
    }

    pool_kernel<<<(BATCH * DMODEL + 255) / 256, 256, 0, stream>>>(
        H, POOLED, BATCH, SEQ, DMODEL);

    gemm(POOLED, dec1_w, dec1_b, T1, BATCH, DMODEL / 2, DMODEL, /*ReLU*/1);
    gemm(T1, dec2_w, dec2_b, (float*)d_out, BATCH, NCLS, DMODEL / 2, 0);
}